// GraphPooling_5222680232056
// MI455X (gfx1250) — compile-verified
//
#include <hip/hip_runtime.h>
#include <hip/hip_bf16.h>

typedef __attribute__((ext_vector_type(2))) float v2f;
typedef __attribute__((ext_vector_type(8))) float v8f;

#define N_ROWS 500000
#define C_COLS 256
#define K_TOP  250000
#define PADN   501760      // 245 * 2048, >= N_ROWS
#define RADIX_BLOCKS 245
#define CHUNK  2048

// Map float score -> 32-bit key such that ASCENDING key order == DESCENDING score
// order, and a stable sort then breaks ties by ascending original index (matches
// jax.lax.top_k / torch.topk largest=True).
__device__ __forceinline__ unsigned key_from_score(float s) {
    unsigned u = __float_as_uint(s);
    u ^= (u >> 31) ? 0xFFFFFFFFu : 0x80000000u;   // monotone ascending map
    return ~u;                                    // invert for descending-score
}

// ---------------------------------------------------------------------------
// Pad region [N_ROWS, PADN): max keys sort to the end, never enter top-K.
__global__ void pad_kernel(unsigned* __restrict__ keys, unsigned* __restrict__ idx) {
    int i = blockIdx.x * blockDim.x + threadIdx.x;
    int e = N_ROWS + i;
    if (e < PADN) { keys[e] = 0xFFFFFFFFu; idx[e] = (unsigned)e; }
}

// ---------------------------------------------------------------------------
// GEMV via V_WMMA_F32_16X16X4_F32. One wave -> 16 rows of x.
// A (16x4 f32): lane L holds x[row = L%16][k = 4t + 2*(L/16) + {0,1}]
// B (4x16 f32): broadcast W[4t + k] across all 16 columns -> every column of D
//               is the same dot product; accumulate over t = 0..63.
// D (16x16 f32): lane 0 VGPR p = D[p][0], lane 16 VGPR p = D[8+p][0].
__global__ void scores_wmma_kernel(const float* __restrict__ x,
                                   const float* __restrict__ W,
                                   const float* __restrict__ bptr,
                                   unsigned* __restrict__ keys,
                                   unsigned* __restrict__ idx) {
    int tid  = blockIdx.x * blockDim.x + threadIdx.x;
    int wave = tid >> 5;
    int lane = threadIdx.x & 31;
    long rowBase = (long)wave * 16;
    int  m = lane & 15;
    int  h = lane >> 4;                 // 0 -> K {0,1}, 1 -> K {2,3}
    long row = rowBase + m;
    if (row >= N_ROWS) row = N_ROWS - 1;          // clamp loads; writes guarded
    const float* xrow = x + row * (long)C_COLS;

    v8f c = {};
    #pragma unroll 8
    for (int t = 0; t < 64; ++t) {
        v2f a  = *(const v2f*)(xrow + 4 * t + 2 * h);
        v2f bm = *(const v2f*)(W    + 4 * t + 2 * h);
        c = __builtin_amdgcn_wmma_f32_16x16x4_f32(
                /*neg_a=*/false, a, /*neg_b=*/false, bm,
                /*c_mod=*/(short)0, c, /*reuse_a=*/false, /*reuse_b=*/false);
    }

    float bias = *bptr;
    if (lane == 0 || lane == 16) {
        long rb = rowBase + (lane == 16 ? 8 : 0);
        #pragma unroll
        for (int p = 0; p < 8; ++p) {
            long r = rb + p;
            if (r < N_ROWS) {
                keys[r] = key_from_score(c[p] + bias);
                idx[r]  = (unsigned)r;
            }
        }
    }
}

// ---------------------------------------------------------------------------
// Radix pass 1/3: per-block digit histogram. counts[block*256 + digit].
__global__ void hist_kernel(const unsigned* __restrict__ keys,
                            unsigned* __restrict__ counts, int shift) {
    __shared__ unsigned h[256];
    int t = threadIdx.x;
    h[t] = 0;
    __syncthreads();
    long base = (long)blockIdx.x * CHUNK;
    #pragma unroll
    for (int r = 0; r < 8; ++r) {
        unsigned k = keys[base + r * 256 + t];
        atomicAdd(&h[(k >> shift) & 255u], 1u);
    }
    __syncthreads();
    counts[blockIdx.x * 256 + t] = h[t];
}

// Radix pass 2/3: single block. counts -> global scatter base per (block,digit).
__global__ void scan_kernel(unsigned* __restrict__ counts) {
    __shared__ unsigned tot[256];
    __shared__ unsigned scn[256];
    int d = threadIdx.x;
    unsigned run = 0;
    for (int b = 0; b < RADIX_BLOCKS; ++b) {          // exclusive prefix across blocks
        unsigned v = counts[b * 256 + d];
        counts[b * 256 + d] = run;
        run += v;
    }
    tot[d] = run;
    scn[d] = run;
    __syncthreads();
    for (int off = 1; off < 256; off <<= 1) {         // Hillis-Steele inclusive scan
        unsigned add = (d >= off) ? scn[d - off] : 0u;
        __syncthreads();
        scn[d] += add;
        __syncthreads();
    }
    unsigned digitStart = scn[d] - tot[d];            // exclusive digit base
    for (int b = 0; b < RADIX_BLOCKS; ++b)
        counts[b * 256 + d] += digitStart;
}

// Radix pass 3/3: stable scatter. 8 sequential rounds of 256 elements; within a
// round, wave multi-split (ballot) ranks lanes, LDS scan orders the 8 waves.
__global__ void scatter_kernel(const unsigned* __restrict__ keysIn,
                               const unsigned* __restrict__ idxIn,
                               unsigned* __restrict__ keysOut,
                               unsigned* __restrict__ idxOut,
                               const unsigned* __restrict__ counts, int shift) {
    __shared__ unsigned gbase[256];
    __shared__ unsigned whist[8][256];
    __shared__ unsigned rtot[256];
    int t    = threadIdx.x;
    int lane = t & 31;
    int w    = t >> 5;
    gbase[t] = counts[blockIdx.x * 256 + t];
    long base = (long)blockIdx.x * CHUNK;
    unsigned long long lanemask_lt = (1ull << lane) - 1ull;

    for (int r = 0; r < 8; ++r) {
        #pragma unroll
        for (int i = 0; i < 8; ++i) whist[i][t] = 0;
        __syncthreads();

        long e = base + r * 256 + t;
        unsigned k   = keysIn[e];
        unsigned id  = idxIn[e];
        unsigned dig = (k >> shift) & 255u;

        unsigned long long mask = 0xFFFFFFFFull;      // wave32 multi-split
        #pragma unroll
        for (int bit = 0; bit < 8; ++bit) {
            unsigned long long vote = __ballot((dig >> bit) & 1u);
            mask &= ((dig >> bit) & 1u) ? vote : ~vote;
        }
        unsigned rankw = (unsigned)__popcll(mask & lanemask_lt);
        unsigned cntw  = (unsigned)__popcll(mask);
        if (rankw == 0) whist[w][dig] = cntw;
        __syncthreads();

        // per-digit exclusive scan across the 8 waves (thread t owns digit t)
        unsigned runw = 0, offs[8];
        #pragma unroll
        for (int i = 0; i < 8; ++i) { offs[i] = runw; runw += whist[i][t]; }
        __syncthreads();
        #pragma unroll
        for (int i = 0; i < 8; ++i) whist[i][t] = offs[i];
        rtot[t] = runw;
        __syncthreads();

        unsigned pos = gbase[dig] + whist[w][dig] + rankw;
        keysOut[pos] = k;
        idxOut[pos]  = id;
        __syncthreads();

        gbase[t] += rtot[t];
        __syncthreads();
    }
}

// ---------------------------------------------------------------------------
// x_pool[k][:] = x[idx[k]][:] with float4; 64 consecutive lanes share one idx.
__global__ void gather_kernel(const float4* __restrict__ x4,
                              const unsigned* __restrict__ idxSorted,
                              float4* __restrict__ out4) {
    long gid = (long)blockIdx.x * blockDim.x + threadIdx.x;
    long k   = gid >> 6;
    int  c4  = (int)(gid & 63);
    if (k < K_TOP) {
        unsigned id = idxSorted[k];
        out4[k * 64 + c4] = x4[(long)id * 64 + c4];
    }
}

__global__ void idx_out_kernel(const unsigned* __restrict__ idxSorted,
                               float* __restrict__ out) {
    int k = blockIdx.x * blockDim.x + threadIdx.x;
    if (k < K_TOP) out[(long)K_TOP * C_COLS + k] = (float)idxSorted[k];
}

// ---------------------------------------------------------------------------
extern "C" void kernel_launch(void* const* d_in, const int* in_sizes, int n_in,
                              void* d_out, int out_size, void* d_ws, size_t ws_size,
                              hipStream_t stream) {
    const float* x = (const float*)d_in[0];
    const float* W = (const float*)d_in[1];
    const float* b = (const float*)d_in[2];
    float* out = (float*)d_out;

    unsigned* keysA = (unsigned*)d_ws;
    unsigned* keysB = keysA + PADN;
    unsigned* idxA  = keysB + PADN;
    unsigned* idxB  = idxA  + PADN;
    unsigned* counts = idxB + PADN;        // RADIX_BLOCKS*256 u32

    pad_kernel<<<(PADN - N_ROWS + 255) / 256, 256, 0, stream>>>(keysA, idxA);

    int nwaves  = N_ROWS / 16;             // 31250 waves, 16 rows each
    int nblocks = (nwaves + 7) / 8;        // 8 waves (256 threads) per block
    scores_wmma_kernel<<<nblocks, 256, 0, stream>>>(x, W, b, keysA, idxA);

    unsigned *ki = keysA, *ii = idxA, *ko = keysB, *io = idxB;
    for (int pass = 0; pass < 4; ++pass) {
        int shift = pass * 8;
        hist_kernel<<<RADIX_BLOCKS, 256, 0, stream>>>(ki, counts, shift);
        scan_kernel<<<1, 256, 0, stream>>>(counts);
        scatter_kernel<<<RADIX_BLOCKS, 256, 0, stream>>>(ki, ii, ko, io, counts, shift);
        unsigned* tk = ki; ki = ko; ko = tk;
        unsigned* ti = ii; ii = io; io = ti;
    }
    // 4 passes -> sorted pairs back in keysA/idxA (== ki/ii)

    gather_kernel<<<(K_TOP * (C_COLS / 4) + 255) / 256, 256, 0, stream>>>(
        (const float4*)x, ii, (float4*)out);
    idx_out_kernel<<<(K_TOP + 255) / 256, 256, 0, stream>>>(ii, out);
}